// MaskedReconstructionLoss_16037407883312
// MI455X (gfx1250) — compile-verified
//
#include <hip/hip_runtime.h>
#include <stdint.h>

// Shapes from reference: (8,1,64,256,256) f32
#define NB 8
#define S_PER 4194304            // 64*256*256 elements per sample
#define NVEC (S_PER / 4)         // float4 per sample

// ---------------- workspace layout (byte offsets) ----------------
static constexpr size_t OFF_HIST1 = 0;                                   // u32 [NB][2048]
static constexpr size_t OFF_HIST2 = OFF_HIST1 + (size_t)NB * 2048 * 4;   // u32 [NB][5][2048]
static constexpr size_t OFF_HIST3 = OFF_HIST2 + (size_t)NB * 5 * 2048 * 4; // u32 [NB][5][1024]
static constexpr size_t OFF_HM1   = OFF_HIST3 + (size_t)NB * 5 * 1024 * 4; // u32 [NB][2048]
static constexpr size_t OFF_HM2   = OFF_HM1 + (size_t)NB * 2048 * 4;       // u32 [NB][2048]
static constexpr size_t OFF_HM3   = OFF_HM2 + (size_t)NB * 2048 * 4;       // u32 [NB][1024]
static constexpr size_t OFF_NVIS  = OFF_HM3 + (size_t)NB * 1024 * 4;     // u32 [NB]
static constexpr size_t OFF_PFX1  = OFF_NVIS + NB * 4;                   // u32 [NB][5]
static constexpr size_t OFF_REM1  = OFF_PFX1 + NB * 5 * 4;
static constexpr size_t OFF_PFX2  = OFF_REM1 + NB * 5 * 4;
static constexpr size_t OFF_REM2  = OFF_PFX2 + NB * 5 * 4;
static constexpr size_t OFF_MED   = OFF_REM2 + NB * 5 * 4;               // f32 [NB]
static constexpr size_t OFF_IQR   = OFF_MED + NB * 4;                    // f32 [NB]
static constexpr size_t OFF_PM1   = OFF_IQR + NB * 4;                    // u32 [NB]
static constexpr size_t OFF_RM1   = OFF_PM1 + NB * 4;
static constexpr size_t OFF_PM2   = OFF_RM1 + NB * 4;
static constexpr size_t OFF_RM2   = OFF_PM2 + NB * 4;
static constexpr size_t OFF_FMAD  = OFF_RM2 + NB * 4;                    // f32 [NB]
static constexpr size_t OFF_DN    = OFF_FMAD + NB * 4;                   // u32 [NB]
static constexpr size_t OFF_LSUM  = (OFF_DN + NB * 4 + 7) & ~(size_t)7; // double
static constexpr size_t OFF_MSUM  = OFF_LSUM + 8;                        // double
static constexpr size_t WS_BYTES  = OFF_MSUM + 8;

// ---------------- helpers ----------------
__device__ __forceinline__ unsigned sortkey(float f) {
    unsigned u = __float_as_uint(f);
    return u ^ ((u >> 31) ? 0xFFFFFFFFu : 0x80000000u);
}
__device__ __forceinline__ float inv_sortkey(unsigned k) {
    unsigned u = (k & 0x80000000u) ? (k ^ 0x80000000u) : ~k;
    return __uint_as_float(u);
}
__device__ __forceinline__ bool visible_of(float m) {
    return fminf(fmaxf(m, 0.0f), 1.0f) < 1.0f;   // visible iff clip(mask) < 1
}

// ================= pass 1: top-11-bit histogram of visible target =================
__global__ void k_hist1(const float* __restrict__ target, const float* __restrict__ mask,
                        char* __restrict__ ws) {
    __shared__ unsigned h[2048];
    const int s = blockIdx.y;
    for (int i = threadIdx.x; i < 2048; i += blockDim.x) h[i] = 0;
    __syncthreads();
    const float4* t4 = (const float4*)(target + (size_t)s * S_PER);
    const float4* m4 = (const float4*)(mask   + (size_t)s * S_PER);
    for (int i = blockIdx.x * blockDim.x + threadIdx.x; i < NVEC; i += gridDim.x * blockDim.x) {
        float4 t = t4[i]; float4 m = m4[i];
        float tv[4] = {t.x, t.y, t.z, t.w};
        float mv[4] = {m.x, m.y, m.z, m.w};
        #pragma unroll
        for (int k = 0; k < 4; ++k)
            if (visible_of(mv[k])) atomicAdd(&h[sortkey(tv[k]) >> 21], 1u);
    }
    __syncthreads();
    unsigned* g = (unsigned*)(ws + OFF_HIST1) + (size_t)s * 2048;
    for (int i = threadIdx.x; i < 2048; i += blockDim.x) {
        unsigned v = h[i];
        if (v) atomicAdd(&g[i], v);
    }
}

// ================= select level 1 (5 ranks per sample) =================
__global__ void k_select1(char* __restrict__ ws) {
    const int tid = threadIdx.x;
    if (tid >= NB * 5) return;
    const int s = tid / 5, j = tid % 5;
    const unsigned* h = (const unsigned*)(ws + OFF_HIST1) + (size_t)s * 2048;
    unsigned n = 0;
    for (int i = 0; i < 2048; ++i) n += h[i];
    if (j == 0) ((unsigned*)(ws + OFF_NVIS))[s] = n;
    unsigned r = 0;
    if (n > 0) {
        float nf = (float)(n - 1);
        if      (j == 0) r = (n - 1) >> 1;
        else if (j == 1) r = (unsigned)floorf(0.25f * nf);
        else if (j == 2) r = (unsigned)ceilf (0.25f * nf);
        else if (j == 3) r = (unsigned)floorf(0.75f * nf);
        else             r = (unsigned)ceilf (0.75f * nf);
        if (r > n - 1) r = n - 1;
    }
    unsigned cum = 0, pfx = 0, rem = 0;
    for (int bin = 0; bin < 2048; ++bin) {
        unsigned c = h[bin];
        if (cum + c > r) { pfx = (unsigned)bin; rem = r - cum; break; }
        cum += c;
    }
    ((unsigned*)(ws + OFF_PFX1))[tid] = pfx;
    ((unsigned*)(ws + OFF_REM1))[tid] = rem;
}

// ================= pass 2: mid-11-bit histograms gated on top prefix =================
__global__ void k_hist2(const float* __restrict__ target, const float* __restrict__ mask,
                        char* __restrict__ ws) {
    __shared__ unsigned h[5][2048];
    __shared__ unsigned pfx[5];
    const int s = blockIdx.y;
    for (int i = threadIdx.x; i < 5 * 2048; i += blockDim.x) ((unsigned*)h)[i] = 0;
    if (threadIdx.x < 5) pfx[threadIdx.x] = ((unsigned*)(ws + OFF_PFX1))[s * 5 + threadIdx.x];
    __syncthreads();
    const float4* t4 = (const float4*)(target + (size_t)s * S_PER);
    const float4* m4 = (const float4*)(mask   + (size_t)s * S_PER);
    for (int i = blockIdx.x * blockDim.x + threadIdx.x; i < NVEC; i += gridDim.x * blockDim.x) {
        float4 t = t4[i]; float4 m = m4[i];
        float tv[4] = {t.x, t.y, t.z, t.w};
        float mv[4] = {m.x, m.y, m.z, m.w};
        #pragma unroll
        for (int k = 0; k < 4; ++k) {
            if (!visible_of(mv[k])) continue;
            unsigned key = sortkey(tv[k]);
            unsigned top = key >> 21, mid = (key >> 10) & 2047u;
            #pragma unroll
            for (int j = 0; j < 5; ++j)
                if (top == pfx[j]) atomicAdd(&h[j][mid], 1u);
        }
    }
    __syncthreads();
    unsigned* g = (unsigned*)(ws + OFF_HIST2) + (size_t)s * 5 * 2048;
    for (int i = threadIdx.x; i < 5 * 2048; i += blockDim.x) {
        unsigned v = ((unsigned*)h)[i];
        if (v) atomicAdd(&g[i], v);
    }
}

__global__ void k_select2(char* __restrict__ ws) {
    const int tid = threadIdx.x;
    if (tid >= NB * 5) return;
    const unsigned* h = (const unsigned*)(ws + OFF_HIST2) + (size_t)tid * 2048;
    unsigned r = ((unsigned*)(ws + OFF_REM1))[tid];
    unsigned p1 = ((unsigned*)(ws + OFF_PFX1))[tid];
    unsigned cum = 0, b = 0, rem = 0;
    for (int bin = 0; bin < 2048; ++bin) {
        unsigned c = h[bin];
        if (cum + c > r) { b = (unsigned)bin; rem = r - cum; break; }
        cum += c;
    }
    ((unsigned*)(ws + OFF_PFX2))[tid] = (p1 << 11) | b;
    ((unsigned*)(ws + OFF_REM2))[tid] = rem;
}

// ================= pass 3: low-10-bit histograms gated on 22-bit prefix =================
__global__ void k_hist3(const float* __restrict__ target, const float* __restrict__ mask,
                        char* __restrict__ ws) {
    __shared__ unsigned h[5][1024];
    __shared__ unsigned pfx[5];
    const int s = blockIdx.y;
    for (int i = threadIdx.x; i < 5 * 1024; i += blockDim.x) ((unsigned*)h)[i] = 0;
    if (threadIdx.x < 5) pfx[threadIdx.x] = ((unsigned*)(ws + OFF_PFX2))[s * 5 + threadIdx.x];
    __syncthreads();
    const float4* t4 = (const float4*)(target + (size_t)s * S_PER);
    const float4* m4 = (const float4*)(mask   + (size_t)s * S_PER);
    for (int i = blockIdx.x * blockDim.x + threadIdx.x; i < NVEC; i += gridDim.x * blockDim.x) {
        float4 t = t4[i]; float4 m = m4[i];
        float tv[4] = {t.x, t.y, t.z, t.w};
        float mv[4] = {m.x, m.y, m.z, m.w};
        #pragma unroll
        for (int k = 0; k < 4; ++k) {
            if (!visible_of(mv[k])) continue;
            unsigned key = sortkey(tv[k]);
            unsigned hi22 = key >> 10, lo = key & 1023u;
            #pragma unroll
            for (int j = 0; j < 5; ++j)
                if (hi22 == pfx[j]) atomicAdd(&h[j][lo], 1u);
        }
    }
    __syncthreads();
    unsigned* g = (unsigned*)(ws + OFF_HIST3) + (size_t)s * 5 * 1024;
    for (int i = threadIdx.x; i < 5 * 1024; i += blockDim.x) {
        unsigned v = ((unsigned*)h)[i];
        if (v) atomicAdd(&g[i], v);
    }
}

// select level 3: recover exact order statistics, compute med / iqr_mad per sample
__global__ void k_select3(char* __restrict__ ws) {
    __shared__ float kv[NB * 5];
    const int tid = threadIdx.x;
    if (tid < NB * 5) {
        const unsigned* h = (const unsigned*)(ws + OFF_HIST3) + (size_t)tid * 1024;
        unsigned r = ((unsigned*)(ws + OFF_REM2))[tid];
        unsigned p2 = ((unsigned*)(ws + OFF_PFX2))[tid];
        unsigned cum = 0, b = 0;
        for (int bin = 0; bin < 1024; ++bin) {
            unsigned c = h[bin];
            if (cum + c > r) { b = (unsigned)bin; break; }
            cum += c;
        }
        kv[tid] = inv_sortkey((p2 << 10) | b);
    }
    __syncthreads();
    if (tid < NB) {
        const int s = tid;
        unsigned n = ((unsigned*)(ws + OFF_NVIS))[s];
        float med = 0.0f, iqr = 1e-6f;
        if (n > 0) {
            med = kv[s * 5 + 0];
            float nf = (float)(n - 1);
            float p25 = 0.25f * nf; float f25 = p25 - floorf(p25);
            float q25 = kv[s * 5 + 1] + (kv[s * 5 + 2] - kv[s * 5 + 1]) * f25;
            float p75 = 0.75f * nf; float f75 = p75 - floorf(p75);
            float q75 = kv[s * 5 + 3] + (kv[s * 5 + 4] - kv[s * 5 + 3]) * f75;
            iqr = fmaxf((q75 - q25) / 1.35f, 1e-6f);
        }
        ((float*)(ws + OFF_MED))[s] = med;  // already 0 when n==0
        ((float*)(ws + OFF_IQR))[s] = iqr;
    }
}

// ================= MAD radix-select on |t - med| (3 levels, 1 rank/sample) ==========
__global__ void k_histM1(const float* __restrict__ target, const float* __restrict__ mask,
                         char* __restrict__ ws) {
    __shared__ unsigned h[2048];
    const int s = blockIdx.y;
    const float med = ((const float*)(ws + OFF_MED))[s];
    for (int i = threadIdx.x; i < 2048; i += blockDim.x) h[i] = 0;
    __syncthreads();
    const float4* t4 = (const float4*)(target + (size_t)s * S_PER);
    const float4* m4 = (const float4*)(mask   + (size_t)s * S_PER);
    for (int i = blockIdx.x * blockDim.x + threadIdx.x; i < NVEC; i += gridDim.x * blockDim.x) {
        float4 t = t4[i]; float4 m = m4[i];
        float tv[4] = {t.x, t.y, t.z, t.w};
        float mv[4] = {m.x, m.y, m.z, m.w};
        #pragma unroll
        for (int k = 0; k < 4; ++k)
            if (visible_of(mv[k]))
                atomicAdd(&h[__float_as_uint(fabsf(tv[k] - med)) >> 21], 1u);
    }
    __syncthreads();
    unsigned* g = (unsigned*)(ws + OFF_HM1) + (size_t)s * 2048;
    for (int i = threadIdx.x; i < 2048; i += blockDim.x) {
        unsigned v = h[i];
        if (v) atomicAdd(&g[i], v);
    }
}

__global__ void k_selectM1(char* __restrict__ ws) {
    const int s = threadIdx.x;
    if (s >= NB) return;
    const unsigned* h = (const unsigned*)(ws + OFF_HM1) + (size_t)s * 2048;
    unsigned n = ((unsigned*)(ws + OFF_NVIS))[s];
    unsigned r = (n > 0) ? ((n - 1) >> 1) : 0;
    unsigned cum = 0, pfx = 0, rem = 0;
    for (int bin = 0; bin < 2048; ++bin) {
        unsigned c = h[bin];
        if (cum + c > r) { pfx = (unsigned)bin; rem = r - cum; break; }
        cum += c;
    }
    ((unsigned*)(ws + OFF_PM1))[s] = pfx;
    ((unsigned*)(ws + OFF_RM1))[s] = rem;
}

__global__ void k_histM2(const float* __restrict__ target, const float* __restrict__ mask,
                         char* __restrict__ ws) {
    __shared__ unsigned h[2048];
    const int s = blockIdx.y;
    const float med = ((const float*)(ws + OFF_MED))[s];
    const unsigned pfx = ((const unsigned*)(ws + OFF_PM1))[s];
    for (int i = threadIdx.x; i < 2048; i += blockDim.x) h[i] = 0;
    __syncthreads();
    const float4* t4 = (const float4*)(target + (size_t)s * S_PER);
    const float4* m4 = (const float4*)(mask   + (size_t)s * S_PER);
    for (int i = blockIdx.x * blockDim.x + threadIdx.x; i < NVEC; i += gridDim.x * blockDim.x) {
        float4 t = t4[i]; float4 m = m4[i];
        float tv[4] = {t.x, t.y, t.z, t.w};
        float mv[4] = {m.x, m.y, m.z, m.w};
        #pragma unroll
        for (int k = 0; k < 4; ++k) {
            if (!visible_of(mv[k])) continue;
            unsigned key = __float_as_uint(fabsf(tv[k] - med));
            if ((key >> 21) == pfx) atomicAdd(&h[(key >> 10) & 2047u], 1u);
        }
    }
    __syncthreads();
    unsigned* g = (unsigned*)(ws + OFF_HM2) + (size_t)s * 2048;
    for (int i = threadIdx.x; i < 2048; i += blockDim.x) {
        unsigned v = h[i];
        if (v) atomicAdd(&g[i], v);
    }
}

__global__ void k_selectM2(char* __restrict__ ws) {
    const int s = threadIdx.x;
    if (s >= NB) return;
    const unsigned* h = (const unsigned*)(ws + OFF_HM2) + (size_t)s * 2048;
    unsigned r = ((unsigned*)(ws + OFF_RM1))[s];
    unsigned p1 = ((unsigned*)(ws + OFF_PM1))[s];
    unsigned cum = 0, b = 0, rem = 0;
    for (int bin = 0; bin < 2048; ++bin) {
        unsigned c = h[bin];
        if (cum + c > r) { b = (unsigned)bin; rem = r - cum; break; }
        cum += c;
    }
    ((unsigned*)(ws + OFF_PM2))[s] = (p1 << 11) | b;
    ((unsigned*)(ws + OFF_RM2))[s] = rem;
}

__global__ void k_histM3(const float* __restrict__ target, const float* __restrict__ mask,
                         char* __restrict__ ws) {
    __shared__ unsigned h[1024];
    const int s = blockIdx.y;
    const float med = ((const float*)(ws + OFF_MED))[s];
    const unsigned pfx = ((const unsigned*)(ws + OFF_PM2))[s];
    for (int i = threadIdx.x; i < 1024; i += blockDim.x) h[i] = 0;
    __syncthreads();
    const float4* t4 = (const float4*)(target + (size_t)s * S_PER);
    const float4* m4 = (const float4*)(mask   + (size_t)s * S_PER);
    for (int i = blockIdx.x * blockDim.x + threadIdx.x; i < NVEC; i += gridDim.x * blockDim.x) {
        float4 t = t4[i]; float4 m = m4[i];
        float tv[4] = {t.x, t.y, t.z, t.w};
        float mv[4] = {m.x, m.y, m.z, m.w};
        #pragma unroll
        for (int k = 0; k < 4; ++k) {
            if (!visible_of(mv[k])) continue;
            unsigned key = __float_as_uint(fabsf(tv[k] - med));
            if ((key >> 10) == pfx) atomicAdd(&h[key & 1023u], 1u);
        }
    }
    __syncthreads();
    unsigned* g = (unsigned*)(ws + OFF_HM3) + (size_t)s * 1024;
    for (int i = threadIdx.x; i < 1024; i += blockDim.x) {
        unsigned v = h[i];
        if (v) atomicAdd(&g[i], v);
    }
}

__global__ void k_selectM3(char* __restrict__ ws) {
    const int s = threadIdx.x;
    if (s >= NB) return;
    const unsigned* h = (const unsigned*)(ws + OFF_HM3) + (size_t)s * 1024;
    unsigned r = ((unsigned*)(ws + OFF_RM2))[s];
    unsigned p2 = ((unsigned*)(ws + OFF_PM2))[s];
    unsigned cum = 0, b = 0;
    for (int bin = 0; bin < 1024; ++bin) {
        unsigned c = h[bin];
        if (cum + c > r) { b = (unsigned)bin; break; }
        cum += c;
    }
    float mad = __uint_as_float((p2 << 10) | b);
    unsigned n = ((unsigned*)(ws + OFF_NVIS))[s];
    float scaled = 1.4826f * mad;
    float fm = (scaled < 0.1f) ? ((float*)(ws + OFF_IQR))[s] : scaled;
    if (n == 0) fm = 1.0f;
    ((float*)(ws + OFF_FMAD))[s] = fm;
    ((unsigned*)(ws + OFF_DN))[s] = (n > 0 && fm > 0.1f) ? 1u : 0u;
}

// ================= final masked-MSE pass: async global->LDS double buffering =========
__global__ void k_loss(const float* __restrict__ pred, const float* __restrict__ target,
                       const float* __restrict__ mask, char* __restrict__ ws) {
    // per-wave staging: [wave][buf][tensor][128 floats] = 24 KB
    __shared__ float stage[8][2][3][128];
    __shared__ double red[256];
    const int s = blockIdx.y;
    const float* ps = pred   + (size_t)s * S_PER;
    const float* ts = target + (size_t)s * S_PER;
    const float* ms = mask   + (size_t)s * S_PER;
    const float med = ((const float*)(ws + OFF_MED))[s];
    const float fm  = ((const float*)(ws + OFF_FMAD))[s];
    const unsigned dn = ((const unsigned*)(ws + OFF_DN))[s];

    const int wave = threadIdx.x >> 5;
    const int lane = threadIdx.x & 31;
    const int NT = S_PER / 128;          // 32768 tiles of 128 floats
    const int stride = gridDim.x * 8;    // waves in grid (per sample)
    const int tile0 = blockIdx.x * 8 + wave;

    double lsum = 0.0, msum = 0.0;

    auto prefetch = [&](int tile, int b) {
        unsigned goff = (unsigned)(tile * 512 + lane * 16);
        unsigned lp = (unsigned)(uintptr_t)&stage[wave][b][0][lane * 4];
        unsigned lt = (unsigned)(uintptr_t)&stage[wave][b][1][lane * 4];
        unsigned lm = (unsigned)(uintptr_t)&stage[wave][b][2][lane * 4];
        asm volatile("global_load_async_to_lds_b128 %0, %1, %2"
                     :: "v"(lp), "v"(goff), "s"(ps) : "memory");
        asm volatile("global_load_async_to_lds_b128 %0, %1, %2"
                     :: "v"(lt), "v"(goff), "s"(ts) : "memory");
        asm volatile("global_load_async_to_lds_b128 %0, %1, %2"
                     :: "v"(lm), "v"(goff), "s"(ms) : "memory");
    };

    int buf = 0;
    if (tile0 < NT) prefetch(tile0, buf);
    for (int tile = tile0; tile < NT; tile += stride) {
        int nxt = tile + stride;
        if (nxt < NT) {
            prefetch(nxt, buf ^ 1);
            asm volatile("s_wait_asynccnt 0x3" ::: "memory");   // oldest 3 (this tile) done
        } else {
            asm volatile("s_wait_asynccnt 0x0" ::: "memory");
        }
        float4 p4 = *(const float4*)&stage[wave][buf][0][lane * 4];
        float4 t4 = *(const float4*)&stage[wave][buf][1][lane * 4];
        float4 m4 = *(const float4*)&stage[wave][buf][2][lane * 4];
        float pv[4] = {p4.x, p4.y, p4.z, p4.w};
        float tv[4] = {t4.x, t4.y, t4.z, t4.w};
        float mv[4] = {m4.x, m4.y, m4.z, m4.w};
        #pragma unroll
        for (int k = 0; k < 4; ++k) {
            float mc = fminf(fmaxf(mv[k], 0.0f), 1.0f);
            float t = tv[k], p = pv[k];
            if (dn) { t = (t - med) / fm; p = (p - med) / fm; }
            float d = p - t;
            float l = d * d;
            if (!isfinite(l)) l = 100.0f;
            l = fminf(fmaxf(l, 0.0f), 100.0f);
            lsum += (double)(l * mc);
            msum += (double)mc;
        }
        buf ^= 1;
    }

    red[threadIdx.x] = lsum; __syncthreads();
    for (int off = 128; off > 0; off >>= 1) {
        if (threadIdx.x < off) red[threadIdx.x] += red[threadIdx.x + off];
        __syncthreads();
    }
    if (threadIdx.x == 0) atomicAdd((double*)(ws + OFF_LSUM), red[0]);
    __syncthreads();
    red[threadIdx.x] = msum; __syncthreads();
    for (int off = 128; off > 0; off >>= 1) {
        if (threadIdx.x < off) red[threadIdx.x] += red[threadIdx.x + off];
        __syncthreads();
    }
    if (threadIdx.x == 0) atomicAdd((double*)(ws + OFF_MSUM), red[0]);
}

__global__ void k_final(const char* __restrict__ ws, float* __restrict__ out) {
    double ls = *(const double*)(ws + OFF_LSUM);
    double msd = *(const double*)(ws + OFF_MSUM);
    out[0] = (msd > 0.0) ? (float)(ls / fmax(msd, 1.0)) : 0.0f;
}

// ================= host launcher =================
extern "C" void kernel_launch(void* const* d_in, const int* in_sizes, int n_in,
                              void* d_out, int out_size, void* d_ws, size_t ws_size,
                              hipStream_t stream) {
    const float* pred   = (const float*)d_in[0];
    const float* target = (const float*)d_in[1];
    const float* mask   = (const float*)d_in[2];
    char* ws = (char*)d_ws;
    float* out = (float*)d_out;

    hipMemsetAsync(ws, 0, WS_BYTES, stream);

    dim3 hb(256);
    dim3 hg(256, NB);

    k_hist1  <<<hg, hb, 0, stream>>>(target, mask, ws);
    k_select1<<<1, 64, 0, stream>>>(ws);
    k_hist2  <<<hg, hb, 0, stream>>>(target, mask, ws);
    k_select2<<<1, 64, 0, stream>>>(ws);
    k_hist3  <<<hg, hb, 0, stream>>>(target, mask, ws);
    k_select3<<<1, 64, 0, stream>>>(ws);

    k_histM1  <<<hg, hb, 0, stream>>>(target, mask, ws);
    k_selectM1<<<1, 64, 0, stream>>>(ws);
    k_histM2  <<<hg, hb, 0, stream>>>(target, mask, ws);
    k_selectM2<<<1, 64, 0, stream>>>(ws);
    k_histM3  <<<hg, hb, 0, stream>>>(target, mask, ws);
    k_selectM3<<<1, 64, 0, stream>>>(ws);

    k_loss <<<dim3(128, NB), 256, 0, stream>>>(pred, target, mask, ws);
    k_final<<<1, 1, 0, stream>>>(ws, out);
}